// HetSIREN_60550448939627
// MI455X (gfx1250) — compile-verified
//
#include <hip/hip_runtime.h>
#include <hip/hip_bf16.h>

// ---------------------------------------------------------------------------
// HetSIREN fused pipeline for MI455X (gfx1250, wave32).
//   K0: zero the 8 x 128^3 output grids (64 MB).
//   K1: tiny SIREN MLP -> h[8][8], bf16-rounded, stored zero-padded 16x8.
//   K2: per-wave WMMA f32 16x16x4 tile of  h @ Wf  (16 cols = 4 points),
//       + bias, stage in LDS, trilinear scatter via global f32 atomics.
// Memory-bound: ~32MB Wf + 64MB zero-fill + 16.7M atomics (grids fit in L2).
// ---------------------------------------------------------------------------

#define NPTS   262144
#define NB     8
#define LAT    16
#define VOL    128
#define FOURN  (4 * NPTS)          // 1048576 columns of Wf
#define FACTOR 64.0f

typedef __attribute__((ext_vector_type(2))) float f32x2;
typedef __attribute__((ext_vector_type(8))) float f32x8;

__device__ __forceinline__ float bfr(float v) {
    // round through bfloat16 (mimics reference bf16 dtype)
    return __bfloat162float(__float2bfloat16(v));
}

// ---------------------------------------------------------------------------
// K0: zero output grids
// ---------------------------------------------------------------------------
__global__ void zero_out_kernel(float4* __restrict__ out, int nvec) {
    int i = blockIdx.x * blockDim.x + threadIdx.x;
    if (i < nvec) out[i] = make_float4(0.f, 0.f, 0.f, 0.f);
}

// ---------------------------------------------------------------------------
// K1: SIREN head. 64 threads: thread t -> (batch b = t>>3, col j = t&7).
//     hpad is float[16][8], rows 8..15 zeroed (WMMA A-matrix padding).
// ---------------------------------------------------------------------------
__global__ void siren_head_kernel(const float* __restrict__ x,
                                  const float* __restrict__ W0,
                                  const float* __restrict__ b0,
                                  const float* __restrict__ Wh,
                                  const float* __restrict__ bh,
                                  float* __restrict__ hpad) {
    __shared__ float h[NB][8];
    const int t = threadIdx.x;            // 0..63
    const int b = t >> 3;
    const int j = t & 7;

    // h = sin(30 * (x_bf16 @ W0_bf16 + b0))
    float acc = 0.f;
    #pragma unroll
    for (int k = 0; k < LAT; ++k)
        acc += bfr(x[b * LAT + k]) * bfr(W0[k * 8 + j]);
    h[b][j] = bfr(sinf(30.0f * bfr(bfr(acc) + bfr(b0[j]))));
    __syncthreads();

    // 4 residual sine layers
    for (int i = 0; i < 4; ++i) {
        float a = 0.f;
        #pragma unroll
        for (int k = 0; k < 8; ++k)
            a += h[b][k] * bfr(Wh[(i * 8 + k) * 8 + j]);
        float s = bfr(h[b][j] + bfr(sinf(bfr(a) + bfr(bh[i * 8 + j]))));
        __syncthreads();
        h[b][j] = s;
        __syncthreads();
    }

    hpad[b * 8 + j]  = h[b][j];   // rows 0..7  : h (bf16-rounded, used as f32)
    hpad[64 + t]     = 0.0f;      // rows 8..15 : zero padding
}

// ---------------------------------------------------------------------------
// K2: fused WMMA GEMM + trilinear scatter.
//     One wave = one 16-column tile (j0..j0+15) = 4 points x 8 batches.
// ---------------------------------------------------------------------------
__global__ void gemm_scatter_kernel(const float* __restrict__ Wf,
                                    const float* __restrict__ bfv,
                                    const float* __restrict__ hpad,
                                    const float* __restrict__ refv,
                                    const int*  __restrict__ inds,
                                    float* __restrict__ grids) {
    __shared__ float tile[8][NB][16];     // [wave][batch][col-in-tile]

    const int lane = threadIdx.x & 31;
    const int wave = threadIdx.x >> 5;
    const long chunk = (long)blockIdx.x * 8 + wave;   // 0 .. 65535
    const long j0 = chunk * 16;

    const int n  = lane & 15;             // column within tile / A row
    const int hi = lane >> 4;             // lane half selects K pair
    const long j = j0 + n;

    // ---- A matrix (16x4 f32 layout): lane m=n, VGPR0 = K=kb, VGPR1 = K=kb+1
    const int ka = hi * 2;
    f32x2 A0, A1, B0, B1;
    A0.x = hpad[n * 8 + ka];       A0.y = hpad[n * 8 + ka + 1];
    A1.x = hpad[n * 8 + ka + 4];   A1.y = hpad[n * 8 + ka + 5];

    // ---- B matrix (4x16 f32): row k of Wf at stride FOURN
    const long stride = (long)FOURN;
    B0.x = Wf[(long)(ka + 0) * stride + j];
    B0.y = Wf[(long)(ka + 1) * stride + j];
    B1.x = Wf[(long)(ka + 4) * stride + j];
    B1.y = Wf[(long)(ka + 5) * stride + j];

    // ---- D = A*B + C  (two K=4 steps cover K=8)
    f32x8 c = {};
    c = __builtin_amdgcn_wmma_f32_16x16x4_f32(false, A0, false, B0,
                                              (short)0, c, false, false);
    c = __builtin_amdgcn_wmma_f32_16x16x4_f32(false, A1, false, B1,
                                              (short)0, c, false, false);

    // ---- stage tile: lanes 0..15 hold batches 0..7 in c[0..7]
    const float bias = bfv[j];
    if (hi == 0) {
        #pragma unroll
        for (int r = 0; r < NB; ++r)
            tile[wave][r][n] = c[r] + bias;
    }
    __syncthreads();

    // ---- scatter: lane -> (point tp = lane>>3, batch b = lane&7)
    const int tp = lane >> 3;             // 0..3
    const int b  = lane & 7;              // 0..7
    const long p = (j0 >> 2) + tp;        // global point index

    const float dcx = tile[wave][b][tp * 4 + 0];
    const float dcy = tile[wave][b][tp * 4 + 1];
    const float dcz = tile[wave][b][tp * 4 + 2];
    const float dv  = tile[wave][b][tp * 4 + 3];

    const int i0 = inds[p * 3 + 0];
    const int i1 = inds[p * 3 + 1];
    const int i2 = inds[p * 3 + 2];
    const float val = refv[p] + dv;

    // coords = FACTOR*((ind - FACTOR)/FACTOR + delta) + FACTOR = ind + FACTOR*delta
    const float cx = (float)i2 + FACTOR * dcx;   // comp0 <- inds[:,2]
    const float cy = (float)i1 + FACTOR * dcy;   // comp1 <- inds[:,1]
    const float cz = (float)i0 + FACTOR * dcz;   // comp2 <- inds[:,0]

    const float flx = floorf(cx), fly = floorf(cy), flz = floorf(cz);
    const int bx = (int)flx, by = (int)fly, bz = (int)flz;
    const float fx = cx - flx, fy = cy - fly, fz = cz - flz;
    const float gx = 1.f - fx, gy = 1.f - fy, gz = 1.f - fz;

    const float w[8] = { gx*gy*gz, fx*gy*gz, gx*fy*gz, gx*gy*fz,
                         gx*fy*fz, fx*gy*fz, fx*fy*gz, fx*fy*fz };
    const int offs[8][3] = { {0,0,0},{1,0,0},{0,1,0},{0,0,1},
                             {0,1,1},{1,0,1},{1,1,0},{1,1,1} };

    float* __restrict__ g = grids + (long)b * VOL * VOL * VOL;
    #pragma unroll
    for (int i = 0; i < 8; ++i) {
        const int xx = bx + offs[i][0];
        const int yy = by + offs[i][1];
        const int zz = bz + offs[i][2];
        if ((unsigned)xx < (unsigned)VOL &&
            (unsigned)yy < (unsigned)VOL &&
            (unsigned)zz < (unsigned)VOL) {
            atomicAdd(&g[((long)zz * VOL + yy) * VOL + xx], val * w[i]);
        }
    }
}

// ---------------------------------------------------------------------------
extern "C" void kernel_launch(void* const* d_in, const int* in_sizes, int n_in,
                              void* d_out, int out_size, void* d_ws, size_t ws_size,
                              hipStream_t stream) {
    const float* x    = (const float*)d_in[0];   // (8,16)
    const float* W0   = (const float*)d_in[1];   // (16,8)
    const float* b0   = (const float*)d_in[2];   // (8,)
    const float* Wh   = (const float*)d_in[3];   // (4,8,8)
    const float* bh   = (const float*)d_in[4];   // (4,8)
    const float* Wf   = (const float*)d_in[5];   // (8, 4N)
    const float* bfv  = (const float*)d_in[6];   // (4N,)
    const float* refv = (const float*)d_in[7];   // (N,)
    const int*   inds = (const int*)  d_in[8];   // (N,3)

    float* out  = (float*)d_out;                 // (8,128,128,128)
    float* hpad = (float*)d_ws;                  // 16x8 padded A matrix

    // K0: zero 64 MB of output grids
    const int nvec = out_size >> 2;              // float4 count
    zero_out_kernel<<<(nvec + 255) / 256, 256, 0, stream>>>((float4*)out, nvec);

    // K1: tiny SIREN MLP -> hpad
    siren_head_kernel<<<1, 64, 0, stream>>>(x, W0, b0, Wh, bh, hpad);

    // K2: 65536 column chunks, 8 waves/block -> 8192 blocks of 256 threads
    gemm_scatter_kernel<<<8192, 256, 0, stream>>>(Wf, bfv, hpad, refv, inds, out);
}